// PPMIGCN_17532056502863
// MI455X (gfx1250) — compile-verified
//
#include <hip/hip_runtime.h>

typedef __attribute__((ext_vector_type(2))) float v2f;
typedef __attribute__((ext_vector_type(8))) float v8f;

#define IN_DIM 256
#define H1 512
#define H2 128

// ---------------------------------------------------------------------------
// Degree / normalization kernels
// ---------------------------------------------------------------------------
__global__ void deg_init_kernel(float* __restrict__ deg, int n) {
    int i = blockIdx.x * blockDim.x + threadIdx.x;
    if (i < n) deg[i] = 1.0f;  // self-loop weight
}

__global__ void deg_edge_kernel(float* __restrict__ deg,
                                const int* __restrict__ col,
                                const float* __restrict__ ew, int n_edges) {
    int e = blockIdx.x * blockDim.x + threadIdx.x;
    if (e < n_edges) atomicAdd(&deg[col[e]], ew[e]);
}

__global__ void dinv_kernel(float* __restrict__ deg, int n) {
    int i = blockIdx.x * blockDim.x + threadIdx.x;
    if (i < n) {
        float d = deg[i];
        deg[i] = (d > 0.0f) ? rsqrtf(d) : 0.0f;
    }
}

// ---------------------------------------------------------------------------
// FP32 WMMA GEMM: C[M,N] = A[M,K] * B[K,N]   (all row-major)
// One wave computes one 16x64 strip of C (4 accumulators along N) using
// V_WMMA_F32_16X16X4_F32; the A fragment is shared by 4 WMMAs per k-step,
// cutting loads per WMMA from 4 to 2.5 and giving 4 independent acc chains.
// Requires: M % 16 == 0, N % 64 == 0, K % 4 == 0 (true for this workload).
// ---------------------------------------------------------------------------
__global__ void gemm_wmma_f32_kernel(const float* __restrict__ A,
                                     const float* __restrict__ B,
                                     float* __restrict__ C,
                                     int M, int K, int N) {
    int gtid = blockIdx.x * blockDim.x + threadIdx.x;
    int wave = gtid >> 5;
    int lane = threadIdx.x & 31;
    int ntn  = N >> 6;                 // 64-wide strips along N
    int tm   = wave / ntn;
    int tn   = wave % ntn;
    if (tm * 16 >= M) return;          // wave-uniform guard (EXEC all-1 inside)

    int m0 = tm << 4;
    int n0 = tn << 6;
    int half = lane >> 4;              // 0: lanes 0-15, 1: lanes 16-31
    int l    = lane & 15;

    // A 16x4 fragment: lane half h supplies K = k+2h, k+2h+1 for row m0+l
    const float* arow = A + (size_t)(m0 + l) * K + (half << 1);
    // B 4x16 fragments: lane half h supplies rows K = k+2h, k+2h+1, col n0+l+16*t
    const float* brow = B + (size_t)(half << 1) * N + n0 + l;

    v8f acc0 = {}, acc1 = {}, acc2 = {}, acc3 = {};
    for (int k = 0; k < K; k += 4) {
        v2f a;
        a.x = arow[k];
        a.y = arow[k + 1];
        const float* bp = brow + (size_t)k * N;
        v2f b0, b1, b2, b3;
        b0.x = bp[0];      b0.y = bp[N];
        b1.x = bp[16];     b1.y = bp[N + 16];
        b2.x = bp[32];     b2.y = bp[N + 32];
        b3.x = bp[48];     b3.y = bp[N + 48];
        acc0 = __builtin_amdgcn_wmma_f32_16x16x4_f32(false, a, false, b0,
                                                     (short)0, acc0, false, false);
        acc1 = __builtin_amdgcn_wmma_f32_16x16x4_f32(false, a, false, b1,
                                                     (short)0, acc1, false, false);
        acc2 = __builtin_amdgcn_wmma_f32_16x16x4_f32(false, a, false, b2,
                                                     (short)0, acc2, false, false);
        acc3 = __builtin_amdgcn_wmma_f32_16x16x4_f32(false, a, false, b3,
                                                     (short)0, acc3, false, false);
    }

    // C tiles: VGPR r holds row m0 + r + 8*half, col n0 + l + 16*t
    float* crow = C + (size_t)(m0 + (half << 3)) * N + n0 + l;
#pragma unroll
    for (int r = 0; r < 8; ++r) {
        float* cp = crow + (size_t)r * N;
        cp[0]  = acc0[r];
        cp[16] = acc1[r];
        cp[32] = acc2[r];
        cp[48] = acc3[r];
    }
}

// ---------------------------------------------------------------------------
// agg[i][j] = bias[j] + dinv[i]^2 * xw[i][j]     (self-loop message fused)
// ---------------------------------------------------------------------------
__global__ void agg_init_kernel(float* __restrict__ agg,
                                const float* __restrict__ xw,
                                const float* __restrict__ bias,
                                const float* __restrict__ dinv,
                                int n_nodes, int dim) {
    size_t idx = (size_t)blockIdx.x * blockDim.x + threadIdx.x;
    size_t total = (size_t)n_nodes * dim;
    if (idx >= total) return;
    int i = (int)(idx / dim);
    int j = (int)(idx - (size_t)i * dim);
    float di = dinv[i];
    agg[idx] = bias[j] + di * di * xw[idx];
}

// ---------------------------------------------------------------------------
// One wave per edge: agg[col] += dinv[row]*ew*dinv[col] * xw[row]
// Lane-strided dims -> coalesced global_atomic_add_f32.
// ---------------------------------------------------------------------------
__global__ void edge_scatter_kernel(const float* __restrict__ xw,
                                    float* __restrict__ agg,
                                    const int* __restrict__ row,
                                    const int* __restrict__ col,
                                    const float* __restrict__ ew,
                                    const float* __restrict__ dinv,
                                    int n_edges, int dim) {
    int gtid = blockIdx.x * blockDim.x + threadIdx.x;
    int e    = gtid >> 5;
    int lane = threadIdx.x & 31;
    if (e >= n_edges) return;
    int r = row[e];
    int c = col[e];
    float norm = dinv[r] * ew[e] * dinv[c];
    const float* s = xw + (size_t)r * dim;
    float* d = agg + (size_t)c * dim;
    for (int j = lane; j < dim; j += 32)
        atomicAdd(&d[j], norm * s[j]);
}

// ---------------------------------------------------------------------------
// PReLU in place: x = x >= 0 ? x : alpha * x
// ---------------------------------------------------------------------------
__global__ void prelu_kernel(float* __restrict__ x,
                             const float* __restrict__ alpha, size_t n) {
    size_t idx = (size_t)blockIdx.x * blockDim.x + threadIdx.x;
    if (idx >= n) return;
    float v = x[idx];
    float a = alpha[0];
    x[idx] = (v >= 0.0f) ? v : a * v;
}

// ---------------------------------------------------------------------------
// Launch
// ---------------------------------------------------------------------------
extern "C" void kernel_launch(void* const* d_in, const int* in_sizes, int n_in,
                              void* d_out, int out_size, void* d_ws, size_t ws_size,
                              hipStream_t stream) {
    const float* x     = (const float*)d_in[0];   // [N, 256]
    const float* ea    = (const float*)d_in[1];   // [E]
    const float* W1    = (const float*)d_in[2];   // [256, 512]
    const float* b1    = (const float*)d_in[3];   // [512]
    const float* W2    = (const float*)d_in[4];   // [512, 128]
    const float* b2    = (const float*)d_in[5];   // [128]
    const float* alpha = (const float*)d_in[6];   // [1]
    const int*   eidx  = (const int*)d_in[7];     // [2, E]

    const int n_nodes = in_sizes[0] / IN_DIM;
    const int n_edges = in_sizes[1];
    const int* row = eidx;
    const int* col = eidx + n_edges;

    float* out = (float*)d_out;                   // [N, 128]

    // Workspace layout
    float* xw1  = (float*)d_ws;                       // N * H1
    float* agg1 = xw1  + (size_t)n_nodes * H1;        // N * H1
    float* xw2  = agg1 + (size_t)n_nodes * H1;        // N * H2
    float* dinv = xw2  + (size_t)n_nodes * H2;        // N

    const int BLK = 256;

    // 1. degrees (self-loop weight 1 per node) -> dinv
    deg_init_kernel<<<(n_nodes + BLK - 1) / BLK, BLK, 0, stream>>>(dinv, n_nodes);
    deg_edge_kernel<<<(n_edges + BLK - 1) / BLK, BLK, 0, stream>>>(dinv, col, ea, n_edges);
    dinv_kernel<<<(n_nodes + BLK - 1) / BLK, BLK, 0, stream>>>(dinv, n_nodes);

    // 2. GEMM1: xw1 = x @ W1   [N,256]x[256,512]
    {
        long tiles = (long)(n_nodes / 16) * (H1 / 64);
        long threads = tiles * 32;
        gemm_wmma_f32_kernel<<<(threads + BLK - 1) / BLK, BLK, 0, stream>>>(
            x, W1, xw1, n_nodes, IN_DIM, H1);
    }

    // 3. agg1 = b1 + dinv^2 * xw1 ; scatter edges ; PReLU
    {
        size_t total = (size_t)n_nodes * H1;
        agg_init_kernel<<<(total + BLK - 1) / BLK, BLK, 0, stream>>>(
            agg1, xw1, b1, dinv, n_nodes, H1);
        long threads = (long)n_edges * 32;
        edge_scatter_kernel<<<(threads + BLK - 1) / BLK, BLK, 0, stream>>>(
            xw1, agg1, row, col, ea, dinv, n_edges, H1);
        prelu_kernel<<<(total + BLK - 1) / BLK, BLK, 0, stream>>>(agg1, alpha, total);
    }

    // 4. GEMM2: xw2 = feat1 @ W2   [N,512]x[512,128]
    {
        long tiles = (long)(n_nodes / 16) * (H2 / 64);
        long threads = tiles * 32;
        gemm_wmma_f32_kernel<<<(threads + BLK - 1) / BLK, BLK, 0, stream>>>(
            agg1, W2, xw2, n_nodes, H1, H2);
    }

    // 5. out = b2 + dinv^2 * xw2 ; scatter edges ; PReLU
    {
        size_t total = (size_t)n_nodes * H2;
        agg_init_kernel<<<(total + BLK - 1) / BLK, BLK, 0, stream>>>(
            out, xw2, b2, dinv, n_nodes, H2);
        long threads = (long)n_edges * 32;
        edge_scatter_kernel<<<(threads + BLK - 1) / BLK, BLK, 0, stream>>>(
            xw2, out, row, col, ea, dinv, n_edges, H2);
        prelu_kernel<<<(total + BLK - 1) / BLK, BLK, 0, stream>>>(out, alpha, total);
    }
}